// MaskedGraphAutoencoder_41248865911350
// MI455X (gfx1250) — compile-verified
//
#include <hip/hip_runtime.h>

typedef __attribute__((ext_vector_type(2))) float v2f;
typedef __attribute__((ext_vector_type(8))) float v8f;

#define N_NODES 100000
#define N_EDGES 1600000
#define ROW_TILES (N_NODES / 16)   // 6250, exact — no row guards anywhere

// ---------------------------------------------------------------------------
// degree -> inverse degree
// ---------------------------------------------------------------------------
__global__ void deg_kernel(const long long* __restrict__ dst,
                           float* __restrict__ deg) {
    int e = blockIdx.x * blockDim.x + threadIdx.x;
    if (e < N_EDGES) atomicAdd(&deg[(int)dst[e]], 1.0f);
}

__global__ void inv_deg_kernel(float* __restrict__ deg) {
    int i = blockIdx.x * blockDim.x + threadIdx.x;
    if (i < N_NODES) deg[i] = 1.0f / fmaxf(deg[i], 1.0f);
}

// ---------------------------------------------------------------------------
// zero-pad weights (Fin x Fout -> 64 x 64) and bias (Fout -> 64)
// ---------------------------------------------------------------------------
__global__ void pad_weight(const float* __restrict__ w, float* __restrict__ wp,
                           int Fin, int Fout) {
    int i = blockIdx.x * blockDim.x + threadIdx.x;   // 64*64 elements
    if (i >= 64 * 64) return;
    int k = i >> 6, c = i & 63;
    wp[i] = (k < Fin && c < Fout) ? w[k * Fout + c] : 0.0f;
}

__global__ void pad_bias(const float* __restrict__ b, float* __restrict__ bp,
                         int Fout) {
    int i = threadIdx.x;                             // blockDim = 64
    bp[i] = (i < Fout) ? b[i] : 0.0f;
}

// ---------------------------------------------------------------------------
// edge scatter, vectorized: one thread per (edge, 4-feature chunk).
// float4 gather (global_load_b128), 4 float atomics. msg pitch fixed at 64.
// F must be a multiple of 4.
// ---------------------------------------------------------------------------
template <int F, int PIN>
__global__ void scatter4_kernel(const float* __restrict__ h,
                                const long long* __restrict__ src,
                                const long long* __restrict__ dst,
                                float* __restrict__ msg) {
    constexpr int CH = F / 4;                 // chunks per edge
    long long idx = (long long)blockIdx.x * blockDim.x + threadIdx.x;
    if (idx >= (long long)N_EDGES * CH) return;
    int e = (int)(idx / CH);
    int f = 4 * (int)(idx % CH);
    int s = (int)src[e];
    int d = (int)dst[e];
    const float4 v = *(const float4*)(h + (size_t)s * PIN + f);
    float* mp = msg + (size_t)d * 64 + f;
    atomicAdd(mp + 0, v.x);
    atomicAdd(mp + 1, v.y);
    atomicAdd(mp + 2, v.z);
    atomicAdd(mp + 3, v.w);
}

// scalar scatter for the width-3 latent layer
template <int F, int PIN>
__global__ void scatter_kernel(const float* __restrict__ h,
                               const long long* __restrict__ src,
                               const long long* __restrict__ dst,
                               float* __restrict__ msg) {
    long long idx = (long long)blockIdx.x * blockDim.x + threadIdx.x;
    if (idx >= (long long)N_EDGES * F) return;
    int e = (int)(idx / F);
    int f = (int)(idx % F);
    int s = (int)src[e];
    int d = (int)dst[e];
    atomicAdd(&msg[(size_t)d * 64 + f], h[(size_t)s * PIN + f]);
}

// ---------------------------------------------------------------------------
// fused SAGE dense step: out = relu?((msg*inv_deg)@Wl + h@Wr + b)
// One wave = one 16-row stripe x all output col tiles. No guards in K loop:
// weights/bias zero-padded to 64-wide, inputs padded (or exactly sized).
// V_WMMA_F32_16X16X4_F32:
//   A: lane row = lane%16, VGPR v -> K = v + 2*(lane/16)
//   B: lane col = lane%16, VGPR v -> K = v + 2*(lane/16)
//   D: lane col = lane%16, VGPR v -> row = v + 8*(lane/16)
// ---------------------------------------------------------------------------
template <int KP, int PIN, int POUT, int FOUT, int RELU>
__global__ __launch_bounds__(128) void sage_gemm(
    const float* __restrict__ h_in, const float* __restrict__ msg,
    const float* __restrict__ inv_deg, const float* __restrict__ Wl,
    const float* __restrict__ Wr, const float* __restrict__ bias,
    float* __restrict__ out) {
    constexpr int NCT = (POUT + 15) / 16;   // col tiles to compute/store
    const int lane = threadIdx.x & 31;
    const int tile = blockIdx.x * 4 + (threadIdx.x >> 5);
    if (tile >= ROW_TILES) return;          // wave-uniform
    const int row0 = tile << 4;
    const int m  = lane & 15;
    const int kg = lane >> 4;
    const int row = row0 + m;
    const float inv = inv_deg[row];

    const float* __restrict__ mrow = msg + (size_t)row * 64;
    const float* __restrict__ hrow = h_in + (size_t)row * PIN;

    v8f acc[NCT];
#pragma unroll
    for (int t = 0; t < NCT; ++t) acc[t] = (v8f){};

    // pass 1: (msg * inv_deg) @ Wl
#pragma unroll
    for (int k0 = 0; k0 < KP; k0 += 4) {
        const int kb = k0 + 2 * kg;
        v2f a;
        a[0] = mrow[kb] * inv;
        a[1] = mrow[kb + 1] * inv;
#pragma unroll
        for (int t = 0; t < NCT; ++t) {
            v2f b;
            b[0] = Wl[kb * 64 + t * 16 + m];
            b[1] = Wl[(kb + 1) * 64 + t * 16 + m];
            acc[t] = __builtin_amdgcn_wmma_f32_16x16x4_f32(
                false, a, false, b, (short)0, acc[t], false, false);
        }
    }
    // pass 2: h @ Wr
#pragma unroll
    for (int k0 = 0; k0 < KP; k0 += 4) {
        const int kb = k0 + 2 * kg;
        v2f a;
        a[0] = hrow[kb];
        a[1] = hrow[kb + 1];
#pragma unroll
        for (int t = 0; t < NCT; ++t) {
            v2f b;
            b[0] = Wr[kb * 64 + t * 16 + m];
            b[1] = Wr[(kb + 1) * 64 + t * 16 + m];
            acc[t] = __builtin_amdgcn_wmma_f32_16x16x4_f32(
                false, a, false, b, (short)0, acc[t], false, false);
        }
    }

#pragma unroll
    for (int t = 0; t < NCT; ++t) {
        const int c = t * 16 + m;
        if (c < FOUT) {                     // compile-time trivial when FOUT==POUT==64
            const float bv = bias[c];
#pragma unroll
            for (int v = 0; v < 8; ++v) {
                const int r = row0 + v + 8 * kg;
                float val = acc[t][v] + bv;
                if (RELU) val = fmaxf(val, 0.0f);
                out[(size_t)r * POUT + c] = val;
            }
        }
    }
}

// densify z: pitch-4 padded -> N x 3
__global__ void extract_z(const float* __restrict__ zP, float* __restrict__ z) {
    int i = blockIdx.x * blockDim.x + threadIdx.x;
    if (i >= N_NODES * 3) return;
    int n = i / 3, c = i - n * 3;
    z[i] = zP[n * 4 + c];
}

// ---------------------------------------------------------------------------
// host-side orchestration
// ---------------------------------------------------------------------------
template <int F, int PIN>
static void launch_scatter4(const float* h, const long long* src,
                            const long long* dst, float* msg, hipStream_t s) {
    hipMemsetAsync(msg, 0, (size_t)N_NODES * 64 * sizeof(float), s);
    long long tot = (long long)N_EDGES * (F / 4);
    scatter4_kernel<F, PIN><<<(int)((tot + 255) / 256), 256, 0, s>>>(h, src, dst, msg);
}

template <int F, int PIN>
static void launch_scatter(const float* h, const long long* src,
                           const long long* dst, float* msg, hipStream_t s) {
    hipMemsetAsync(msg, 0, (size_t)N_NODES * 64 * sizeof(float), s);
    long long tot = (long long)N_EDGES * F;
    scatter_kernel<F, PIN><<<(int)((tot + 255) / 256), 256, 0, s>>>(h, src, dst, msg);
}

extern "C" void kernel_launch(void* const* d_in, const int* in_sizes, int n_in,
                              void* d_out, int out_size, void* d_ws, size_t ws_size,
                              hipStream_t stream) {
    const float* x = (const float*)d_in[0];
    const long long* edge = (const long long*)d_in[1];
    const long long* src = edge;
    const long long* dst = edge + N_EDGES;

    const float* P[18];
    for (int i = 0; i < 18; ++i) P[i] = (const float*)d_in[2 + i];
    // per-layer (Fin, Fout)
    const int FI[6] = {40, 64, 64, 3, 64, 64};
    const int FO[6] = {64, 64, 3, 64, 64, 40};

    float* ws  = (float*)d_ws;
    float* inv = ws;                              // N
    float* msg = inv + N_NODES;                   // N*64
    float* hA  = msg + (size_t)N_NODES * 64;      // N*64
    float* hB  = hA  + (size_t)N_NODES * 64;      // N*64
    float* zP  = hB  + (size_t)N_NODES * 64;      // N*4
    float* prm = zP  + (size_t)N_NODES * 4;       // 6 * (4096+4096+64)
    float* WlP[6], *WrP[6], *bP[6];
    for (int i = 0; i < 6; ++i) {
        WlP[i] = prm + i * (64 * 64 * 2 + 64);
        WrP[i] = WlP[i] + 64 * 64;
        bP[i]  = WrP[i] + 64 * 64;
    }

    float* x_rec = (float*)d_out;                       // N*40
    float* z_out = x_rec + (size_t)N_NODES * 40;        // N*3

    // pad all parameters to 64-wide
    for (int i = 0; i < 6; ++i) {
        pad_weight<<<16, 256, 0, stream>>>(P[3 * i],     WlP[i], FI[i], FO[i]);
        pad_weight<<<16, 256, 0, stream>>>(P[3 * i + 1], WrP[i], FI[i], FO[i]);
        pad_bias<<<1, 64, 0, stream>>>(P[3 * i + 2], bP[i], FO[i]);
    }

    // degrees -> inverse degrees
    hipMemsetAsync(inv, 0, (size_t)N_NODES * sizeof(float), stream);
    deg_kernel<<<(N_EDGES + 255) / 256, 256, 0, stream>>>(dst, inv);
    inv_deg_kernel<<<(N_NODES + 255) / 256, 256, 0, stream>>>(inv);

    const int gblocks = (ROW_TILES + 3) / 4;

    // enc1: x(40) -> hA(64), relu
    launch_scatter4<40, 40>(x, src, dst, msg, stream);
    sage_gemm<40, 40, 64, 64, 1><<<gblocks, 128, 0, stream>>>(
        x, msg, inv, WlP[0], WrP[0], bP[0], hA);
    // enc2: hA(64) -> hB(64), relu
    launch_scatter4<64, 64>(hA, src, dst, msg, stream);
    sage_gemm<64, 64, 64, 64, 1><<<gblocks, 128, 0, stream>>>(
        hA, msg, inv, WlP[1], WrP[1], bP[1], hB);
    // enc3: hB(64) -> zP(pitch 4, cols 0..2 valid, col 3 exact zero), no relu
    launch_scatter4<64, 64>(hB, src, dst, msg, stream);
    sage_gemm<64, 64, 4, 4, 0><<<gblocks, 128, 0, stream>>>(
        hB, msg, inv, WlP[2], WrP[2], bP[2], zP);
    // dec1: zP(3, pitch 4) -> hA(64), relu
    launch_scatter<3, 4>(zP, src, dst, msg, stream);
    sage_gemm<4, 4, 64, 64, 1><<<gblocks, 128, 0, stream>>>(
        zP, msg, inv, WlP[3], WrP[3], bP[3], hA);
    // dec2: hA(64) -> hB(64), relu
    launch_scatter4<64, 64>(hA, src, dst, msg, stream);
    sage_gemm<64, 64, 64, 64, 1><<<gblocks, 128, 0, stream>>>(
        hA, msg, inv, WlP[4], WrP[4], bP[4], hB);
    // dec3: hB(64) -> x_rec (dense N x 40 in d_out), no relu
    launch_scatter4<64, 64>(hB, src, dst, msg, stream);
    sage_gemm<64, 64, 40, 40, 0><<<gblocks, 128, 0, stream>>>(
        hB, msg, inv, WlP[5], WrP[5], bP[5], x_rec);

    // densify z into d_out
    extract_z<<<(N_NODES * 3 + 255) / 256, 256, 0, stream>>>(zP, z_out);
}